// GCNNet_82197084111147
// MI455X (gfx1250) — compile-verified
//
#include <hip/hip_runtime.h>

// ---------------------------------------------------------------------------
// GCN 2-layer pipeline for MI455X (gfx1250, wave32).
//  - GEMMs: V_WMMA_F32_16X16X4_F32 (fp32, matches reference precision),
//    128x128 block tile, 32x64 wave tile (8 accumulators),
//    double-buffered LDS fed by GLOBAL_LOAD_ASYNC_TO_LDS_B128 (ASYNCcnt).
//  - Edge aggregation: L2-resident gather + global_atomic_add_f32
//    (h/acc ~51MB each fit in the 192MB L2).
// ---------------------------------------------------------------------------

typedef __attribute__((ext_vector_type(2))) float v2f;
typedef __attribute__((ext_vector_type(8))) float v8f;

#define WMMA_F32_16x16x4(a, b, c) \
  __builtin_amdgcn_wmma_f32_16x16x4_f32(false, (a), false, (b), (short)0, (c), false, false)

// async copy of 16B from global to LDS; tracked by ASYNCcnt
#define ASYNC_LD_B128(ldsoff_u32, gaddr_u64)                      \
  asm volatile("global_load_async_to_lds_b128 %0, %1, off"        \
               :: "v"(ldsoff_u32), "v"(gaddr_u64) : "memory")

// ---------------------------------------------------------------- degree ----
__global__ void k_init_deg(float* __restrict__ deg, int n) {
  int i = blockIdx.x * blockDim.x + threadIdx.x;
  if (i < n) deg[i] = 1.0f;  // self-loop contributes 1 to every node's degree
}

__global__ void k_accum_deg(const int* __restrict__ dst, float* __restrict__ deg, int E) {
  int e = blockIdx.x * blockDim.x + threadIdx.x;
  if (e < E) atomicAdd(&deg[dst[e]], 1.0f);
}

__global__ void k_rsqrt_inplace(float* __restrict__ d, int n) {
  int i = blockIdx.x * blockDim.x + threadIdx.x;
  if (i < n) d[i] = rsqrtf(d[i]);  // deg >= 1 always (self-loops)
}

// ------------------------------------------------------------------ GEMM ----
// C[M,N] = A[M,K] @ B[K,N], row-major fp32. K % 16 == 0, N % 128 == 0.
// Block: 256 threads (8 waves); tile 128x128. Wave (wr,wc): rows wr*32..+31,
// cols wc*64..+63 -> v8f acc[2][4]. K staged in chunks of 16, double-buffered
// LDS filled by async b128 loads (exactly 4 per thread per stage).
#define ASTR 20   // A LDS row stride (floats): b64 frag reads cover all 64 banks
#define BSTR 144  // B LDS row stride (floats): half-waves offset by 32 banks

__launch_bounds__(256)
__global__ void k_gemm_wmma_f32(const float* __restrict__ A,
                                const float* __restrict__ B,
                                float* __restrict__ C,
                                int M, int N, int K) {
  __shared__ float lAs[2][128 * ASTR];  // 2 x 10240 B
  __shared__ float lBs[2][16 * BSTR];   // 2 x  9216 B

  const int t    = threadIdx.x;
  const int lane = t & 31;
  const int wave = t >> 5;
  const int wr   = wave & 3;   // row group of 32
  const int wc   = wave >> 2;  // col group of 64
  const int gm0  = blockIdx.x * 128;
  const int gn0  = blockIdx.y * 128;

  // --- async loader geometry: A tile 128x16 and B tile 16x128 = 512 x 16B
  // segments each; thread t moves segments {t, t+256} of both tiles.
  const int aRow0 = t >> 2, aRow1 = aRow0 + 64;     // A row per segment
  const int aCol  = (t & 3) << 2;                   // 4-float column group
  const int bRow0 = t >> 5, bRow1 = bRow0 + 8;      // B k-row per segment
  const int bCol  = (t & 31) << 2;

  // Clamp OOB A rows instead of predicating: keeps the per-wave async-op
  // count uniform (4/stage) so s_wait_asynccnt 4 drains exactly the previous
  // stage. Clamped rows only pollute C rows >= M, which are never stored.
  const int aR0c = min(gm0 + aRow0, M - 1);
  const int aR1c = min(gm0 + aRow1, M - 1);

  const unsigned long long gA0 =
      (unsigned long long)(uintptr_t)(A + (long long)aR0c * K + aCol);
  const unsigned long long gA1 =
      (unsigned long long)(uintptr_t)(A + (long long)aR1c * K + aCol);
  const unsigned long long gB0 =
      (unsigned long long)(uintptr_t)(B + (long long)bRow0 * N + gn0 + bCol);
  const unsigned long long gB1 =
      (unsigned long long)(uintptr_t)(B + (long long)bRow1 * N + gn0 + bCol);
  const unsigned long long aStep = 16ull * sizeof(float);           // +16 k
  const unsigned long long bStep = 16ull * (unsigned)N * sizeof(float);

  unsigned lA0[2], lA1[2], lB0[2], lB1[2];
#pragma unroll
  for (int bf = 0; bf < 2; ++bf) {
    lA0[bf] = (unsigned)(uintptr_t)&lAs[bf][aRow0 * ASTR + aCol];
    lA1[bf] = (unsigned)(uintptr_t)&lAs[bf][aRow1 * ASTR + aCol];
    lB0[bf] = (unsigned)(uintptr_t)&lBs[bf][bRow0 * BSTR + bCol];
    lB1[bf] = (unsigned)(uintptr_t)&lBs[bf][bRow1 * BSTR + bCol];
  }

#define ISSUE_STAGE(bf, kc)                                   \
  do {                                                        \
    unsigned long long ao = (unsigned long long)(kc) * aStep; \
    unsigned long long bo = (unsigned long long)(kc) * bStep; \
    ASYNC_LD_B128(lA0[bf], gA0 + ao);                         \
    ASYNC_LD_B128(lA1[bf], gA1 + ao);                         \
    ASYNC_LD_B128(lB0[bf], gB0 + bo);                         \
    ASYNC_LD_B128(lB1[bf], gB1 + bo);                         \
  } while (0)

  v8f acc[2][4];
#pragma unroll
  for (int i = 0; i < 2; ++i)
#pragma unroll
    for (int j = 0; j < 4; ++j) acc[i][j] = (v8f){};

  // Per-lane WMMA fragment coords (ISA 7.12.2):
  //   A 16x4 f32: M = lane%16, VGPR pair holds K = 2*(lane/16)+{0,1}
  //   B 4x16 f32: N = lane%16, VGPR pair holds K = 2*(lane/16)+{0,1}
  const int m0    = wr * 32 + (lane & 15);
  const int m1    = m0 + 16;
  const int khalf = (lane >> 4) << 1;  // 0 or 2
  const int nb    = wc * 64 + (lane & 15);

  const int nk = K >> 4;  // K chunks of 16
  ISSUE_STAGE(0, 0);

  for (int kc = 0; kc < nk; ++kc) {
    const int buf = kc & 1;
    if (kc + 1 < nk) {
      ISSUE_STAGE(buf ^ 1, kc + 1);
      asm volatile("s_wait_asynccnt 4" ::: "memory");  // prev stage landed
    } else {
      asm volatile("s_wait_asynccnt 0" ::: "memory");
    }
    __syncthreads();  // publish LDS across waves

    const float* __restrict__ pA = &lAs[buf][0];
    const float* __restrict__ pB = &lBs[buf][0];
#pragma unroll
    for (int kk = 0; kk < 4; ++kk) {
      const int kb = (kk << 2) + khalf;
      v2f a0 = *(const v2f*)&pA[m0 * ASTR + kb];  // single ds_load_b64
      v2f a1 = *(const v2f*)&pA[m1 * ASTR + kb];
      v2f b[4];
#pragma unroll
      for (int j = 0; j < 4; ++j) {
        b[j].x = pB[kb * BSTR + nb + j * 16];
        b[j].y = pB[(kb + 1) * BSTR + nb + j * 16];
      }
#pragma unroll
      for (int j = 0; j < 4; ++j) {
        acc[0][j] = WMMA_F32_16x16x4(a0, b[j], acc[0][j]);
        acc[1][j] = WMMA_F32_16x16x4(a1, b[j], acc[1][j]);
      }
    }
    __syncthreads();  // all waves done reading buf before it is refilled
  }

  // C/D 16x16 f32 layout: VGPR r -> M = r + 8*(lane/16), N = lane%16
#pragma unroll
  for (int i = 0; i < 2; ++i) {
    const int mbase = gm0 + wr * 32 + i * 16 + ((lane >> 4) << 3);
#pragma unroll
    for (int r = 0; r < 8; ++r) {
      int m = mbase + r;
      if (m < M) {
        long long o = (long long)m * N + gn0 + wc * 64 + (lane & 15);
        C[o]      = acc[i][0][r];
        C[o + 16] = acc[i][1][r];
        C[o + 32] = acc[i][2][r];
        C[o + 48] = acc[i][3][r];
      }
    }
  }
#undef ISSUE_STAGE
}

// ------------------------------------------------------------------ misc ----
__global__ void k_zero_f4(float* __restrict__ p, long long n4) {
  long long i = blockIdx.x * (long long)blockDim.x + threadIdx.x;
  if (i < n4) ((float4*)p)[i] = make_float4(0.f, 0.f, 0.f, 0.f);
}

// acc[dst] += dinv[src]*dinv[dst] * h[src] for every (non-self-loop) edge.
__global__ void k_scatter(const int* __restrict__ src,
                          const int* __restrict__ dst,
                          const float* __restrict__ dinv,
                          const float* __restrict__ h,
                          float* __restrict__ acc,
                          int E, int c4log2, int C) {
  long long idx   = blockIdx.x * (long long)blockDim.x + threadIdx.x;
  long long total = (long long)E << c4log2;
  if (idx >= total) return;
  int e = (int)(idx >> c4log2);
  int c = (int)(idx & ((1 << c4log2) - 1)) << 2;
  int s = src[e];
  int d = dst[e];
  float nrm = dinv[s] * dinv[d];
  float4 hv = *(const float4*)(h + (long long)s * C + c);
  float* o  = acc + (long long)d * C + c;
  atomicAdd(o + 0, nrm * hv.x);
  atomicAdd(o + 1, nrm * hv.y);
  atomicAdd(o + 2, nrm * hv.z);
  atomicAdd(o + 3, nrm * hv.w);
}

// acc = relu(acc + dinv[i]^2 * h + bias)  (self-loop folded in), in place
__global__ void k_selfloop_bias_relu(float* __restrict__ acc,
                                     const float* __restrict__ h,
                                     const float* __restrict__ dinv,
                                     const float* __restrict__ bias,
                                     int n, int c4log2, int C) {
  long long idx = blockIdx.x * (long long)blockDim.x + threadIdx.x;
  if (idx >= ((long long)n << c4log2)) return;
  int i = (int)(idx >> c4log2);
  int c = (int)(idx & ((1 << c4log2) - 1)) << 2;
  float w = dinv[i]; w *= w;
  long long off = (long long)i * C + c;
  float4 a  = *(float4*)(acc + off);
  float4 hv = *(const float4*)(h + off);
  float4 bv = *(const float4*)(bias + c);
  a.x = fmaxf(fmaf(w, hv.x, a.x) + bv.x, 0.f);
  a.y = fmaxf(fmaf(w, hv.y, a.y) + bv.y, 0.f);
  a.z = fmaxf(fmaf(w, hv.z, a.z) + bv.z, 0.f);
  a.w = fmaxf(fmaf(w, hv.w, a.w) + bv.w, 0.f);
  *(float4*)(acc + off) = a;
}

// out = bias + dinv[i]^2 * t   (init output incl. self-loop before scatter)
__global__ void k_init_out(float* __restrict__ out,
                           const float* __restrict__ t,
                           const float* __restrict__ dinv,
                           const float* __restrict__ bias,
                           int n, int c4log2, int C) {
  long long idx = blockIdx.x * (long long)blockDim.x + threadIdx.x;
  if (idx >= ((long long)n << c4log2)) return;
  int i = (int)(idx >> c4log2);
  int c = (int)(idx & ((1 << c4log2) - 1)) << 2;
  float w = dinv[i]; w *= w;
  long long off = (long long)i * C + c;
  float4 tv = *(const float4*)(t + off);
  float4 bv = *(const float4*)(bias + c);
  float4 o;
  o.x = fmaf(w, tv.x, bv.x);
  o.y = fmaf(w, tv.y, bv.y);
  o.z = fmaf(w, tv.z, bv.z);
  o.w = fmaf(w, tv.w, bv.w);
  *(float4*)(out + off) = o;
}

// ---------------------------------------------------------------- launch ----
extern "C" void kernel_launch(void* const* d_in, const int* in_sizes, int n_in,
                              void* d_out, int out_size, void* d_ws, size_t ws_size,
                              hipStream_t stream) {
  const float* x  = (const float*)d_in[0];
  const int*   ei = (const int*)d_in[1];   // edge_index [2,E] (int32: JAX x64 off)
  const float* W1 = (const float*)d_in[2];
  const float* b1 = (const float*)d_in[3];
  const float* W2 = (const float*)d_in[4];
  const float* b2 = (const float*)d_in[5];
  float* out = (float*)d_out;

  const int K1 = 512, C1 = 256, C2 = 128;
  const int n = in_sizes[0] / K1;   // 50000
  const int E = in_sizes[1] / 2;    // 1600000
  const int* srcs = ei;
  const int* dsts = ei + E;

  // workspace (~103 MB): dinv n | h1 n*C1 | acc1 n*C1 ; t2 reuses h1's slot
  float* ws   = (float*)d_ws;
  float* dinv = ws;
  float* h1   = ws + (((long long)n + 255) & ~255LL);
  float* acc1 = h1 + (long long)n * C1;
  float* t2   = h1;  // h1 dead after k_selfloop_bias_relu (stream ordering)

  const int TPB = 256;

  // 1) symmetric normalization: deg (incl. self-loop) -> dinv = deg^-1/2
  k_init_deg<<<(n + TPB - 1) / TPB, TPB, 0, stream>>>(dinv, n);
  k_accum_deg<<<(E + TPB - 1) / TPB, TPB, 0, stream>>>(dsts, dinv, E);
  k_rsqrt_inplace<<<(n + TPB - 1) / TPB, TPB, 0, stream>>>(dinv, n);

  // 2) h1 = x @ W1   [n,512]x[512,256]
  {
    dim3 grid((n + 127) / 128, C1 / 128);
    k_gemm_wmma_f32<<<grid, TPB, 0, stream>>>(x, W1, h1, n, C1, K1);
  }

  // 3) acc1 = 0 ; edge scatter (C1) ; fused self-loop+bias+relu
  {
    long long n4 = (long long)n * C1 / 4;
    k_zero_f4<<<(unsigned)((n4 + TPB - 1) / TPB), TPB, 0, stream>>>(acc1, n4);
    long long work = (long long)E << 6;  // C1/4 = 64 float4 per edge
    k_scatter<<<(unsigned)((work + TPB - 1) / TPB), TPB, 0, stream>>>(
        srcs, dsts, dinv, h1, acc1, E, 6, C1);
    long long w2 = (long long)n << 6;
    k_selfloop_bias_relu<<<(unsigned)((w2 + TPB - 1) / TPB), TPB, 0, stream>>>(
        acc1, h1, dinv, b1, n, 6, C1);
  }

  // 4) t2 = hrelu @ W2   [n,256]x[256,128]
  {
    dim3 grid((n + 127) / 128, C2 / 128);
    k_gemm_wmma_f32<<<grid, TPB, 0, stream>>>(acc1, W2, t2, n, C2, C1);
  }

  // 5) out = b2 + selfloop, then edge scatter (C2) into d_out
  {
    long long w3 = (long long)n << 5;  // C2/4 = 32 float4 per node
    k_init_out<<<(unsigned)((w3 + TPB - 1) / TPB), TPB, 0, stream>>>(
        out, t2, dinv, b2, n, 5, C2);
    long long work = (long long)E << 5;
    k_scatter<<<(unsigned)((work + TPB - 1) / TPB), TPB, 0, stream>>>(
        srcs, dsts, dinv, t2, out, E, 5, C2);
  }
}